// NMSWrapper_19035295055970
// MI455X (gfx1250) — compile-verified
//
#include <hip/hip_runtime.h>
#include <hip/hip_bf16.h>
#include <stdint.h>

// ---------------- problem constants (YOLOv8-pose @640) ----------------
#define N_ANCH   8400
#define N_PAD    8448            // 33 * 256
#define KD       51
#define MAX_DET  300
#define BLOCK    256
#define KPT      33              // candidates per thread (N_PAD / BLOCK)
#define SCORE_T  0.001f
#define IOU_T    0.7f
#define NEG_INF  (-__builtin_inff())

typedef unsigned int v4u __attribute__((ext_vector_type(4)));
typedef int          v8i __attribute__((ext_vector_type(8)));
typedef int          v4i __attribute__((ext_vector_type(4)));

__global__ __launch_bounds__(BLOCK) void nms_pose_kernel(
    const float* __restrict__ boxes,    // [B, N, 4]
    const float* __restrict__ scores,   // [B, N, 1]
    const float* __restrict__ kpts,     // [B, N, 51]
    float* __restrict__ out_boxes,      // [B, 300, 4]
    float* __restrict__ out_scores,     // [B, 300]
    float* __restrict__ out_labels,     // [B, 300]  (always 0 here, C==1)
    float* __restrict__ out_kpts)       // [B, 300, 51]
{
  __shared__ float4 shBox[N_PAD];      // 135168 B of LDS (WGP has 320KB)
  __shared__ float  wS[BLOCK / 32];
  __shared__ int    wI[BLOCK / 32];
  __shared__ float  bSsh;
  __shared__ int    bIsh;

  const int b    = blockIdx.x;
  const int tid  = threadIdx.x;
  const int lane = tid & 31;
  const int wave = tid >> 5;

  const float* __restrict__ gBox = boxes  + (size_t)b * N_ANCH * 4;
  const float* __restrict__ gScr = scores + (size_t)b * N_ANCH;
  const float* __restrict__ gKpt = kpts   + (size_t)b * N_ANCH * KD;

  float* __restrict__ oB = out_boxes  + (size_t)b * MAX_DET * 4;
  float* __restrict__ oS = out_scores + (size_t)b * MAX_DET;
  float* __restrict__ oL = out_labels + (size_t)b * MAX_DET;
  float* __restrict__ oK = out_kpts   + (size_t)b * MAX_DET * KD;

  // ---------------- stage boxes into LDS ----------------
  // zero the padded tail (indices 8400..8447): degenerate boxes -> IoU 0
  if (tid < (N_PAD - N_ANCH)) {
    shBox[N_ANCH + tid] = make_float4(0.f, 0.f, 0.f, 0.f);
  }

#if __has_builtin(__builtin_amdgcn_tensor_load_to_lds)
  // CDNA5 Tensor Data Mover: one DMA of the 8400x4 f32 tile -> LDS.
  // Issued by wave 0 only (TDM ignores EXEC; one issue per wave).
  if (wave == 0) {
    const uint32_t lds_addr = (uint32_t)(uintptr_t)(&shBox[0]); // low 32b of flat LDS ptr = LDS offset
    const uint64_t ga       = (uint64_t)(uintptr_t)gBox;
    const uint32_t elems    = (uint32_t)(N_ANCH * 4);           // 33600 f32 elements, one row

    v4u g0;
    g0[0] = 1u;                                         // count=1, user mode, no gather
    g0[1] = lds_addr;                                   // lds_addr (bytes)
    g0[2] = (uint32_t)ga;                               // global_addr[31:0]
    g0[3] = (uint32_t)((ga >> 32) & 0x01FFFFFFull) | (2u << 30); // addr[56:32] | type=2

    v8i g1;
    g1[0] = (int)(2u << 16);                            // data_size=2 (4 bytes), no multicast
    g1[1] = (int)((elems & 0xFFFFu) << 16);             // tensor_dim0[15:0]
    g1[2] = (int)(((elems >> 16) & 0xFFFFu) | (1u << 16)); // tensor_dim0[31:16] | tensor_dim1=1
    g1[3] = (int)((elems & 0xFFFFu) << 16);             // tile_dim0 = 33600
    g1[4] = 1;                                          // tile_dim1 = 1, tile_dim2 = 0
    g1[5] = (int)elems;                                 // tensor_dim0_stride[31:0]
    g1[6] = (int)((elems & 0xFFFFu) << 16);             // stride0[47:32]=0 | stride1[15:0]
    g1[7] = 0;                                          // stride1[47:16]

    v4i gz4 = {0, 0, 0, 0};                             // groups 2/3 unused (<=2D tensor)
    v8i gz8 = {0, 0, 0, 0, 0, 0, 0, 0};                 // extra group (clang-23 6-arg form)
    __builtin_amdgcn_tensor_load_to_lds(g0, g1, gz4, gz4, gz8, 0);
    __builtin_amdgcn_s_wait_tensorcnt(0);               // TENSORcnt==0 before sharing LDS
  }
#else
  for (int i = tid; i < N_ANCH; i += BLOCK) {
    shBox[i] = ((const float4*)gBox)[i];
  }
#endif
  __syncthreads();

  // ---------------- per-thread candidate state (registers) ----------------
  // ownership: candidate idx = j*256 + tid  (coalesced, interleaved)
  float rs[KPT];   // live score, -inf = dead/below-threshold
  float ar[KPT];   // precomputed areas
#pragma unroll
  for (int j = 0; j < KPT; ++j) {
    const int idx = j * BLOCK + tid;
    float s = NEG_INF;
    float a = 0.f;
    if (idx < N_ANCH) {
      const float sv = gScr[idx];
      s = (sv > SCORE_T) ? sv : NEG_INF;
      const float4 bb = shBox[idx];
      a = __fmul_rn(__fsub_rn(bb.z, bb.x), __fsub_rn(bb.w, bb.y));
    }
    rs[j] = s;
    ar[j] = a;
  }

  // ---------------- greedy NMS: 300 sequential selections ----------------
  for (int det = 0; det < MAX_DET; ++det) {
    // thread-local argmax (first-index tie-break: strict '>' scanning j asc)
    float ms = rs[0];
    int   mi = tid;            // idx of j=0 is tid
#pragma unroll
    for (int j = 1; j < KPT; ++j) {
      const int idx = j * BLOCK + tid;
      if (rs[j] > ms) { ms = rs[j]; mi = idx; }
    }
    // wave32 reduction, tie-break by lowest global index (matches jnp.argmax)
#pragma unroll
    for (int off = 16; off > 0; off >>= 1) {
      const float os = __shfl_down(ms, off, 32);
      const int   oi = __shfl_down(mi, off, 32);
      if (os > ms || (os == ms && oi < mi)) { ms = os; mi = oi; }
    }
    if (lane == 0) { wS[wave] = ms; wI[wave] = mi; }
    __syncthreads();
    if (tid == 0) {
      float bs = wS[0];
      int   bi = wI[0];
#pragma unroll
      for (int w = 1; w < BLOCK / 32; ++w) {
        const float os = wS[w];
        const int   oi = wI[w];
        if (os > bs || (os == bs && oi < bi)) { bs = os; bi = oi; }
      }
      bSsh = bs; bIsh = bi;
    }
    __syncthreads();

    const float bestS = bSsh;
    const int   bestI = bIsh;
    const bool  valid = (bestS > NEG_INF);   // isfinite(sel_s); scores never +inf/NaN

    // broadcast best box via LDS (all lanes same address -> broadcast read)
    const float4 bb = shBox[bestI];
    const float bArea = __fmul_rn(__fsub_rn(bb.z, bb.x), __fsub_rn(bb.w, bb.y));

    if (valid) {
#pragma unroll
      for (int j = 0; j < KPT; ++j) {
        if (rs[j] != NEG_INF) {            // dead lanes skip LDS load + math
          const int idx = j * BLOCK + tid;
          const float4 c = shBox[idx];
          const float x1 = fmaxf(bb.x, c.x);
          const float y1 = fmaxf(bb.y, c.y);
          const float x2 = fminf(bb.z, c.z);
          const float y2 = fminf(bb.w, c.w);
          const float cw = fmaxf(__fsub_rn(x2, x1), 0.f);
          const float ch = fmaxf(__fsub_rn(y2, y1), 0.f);
          const float inter = __fmul_rn(cw, ch);
          // ((areaA + areaB) - inter) + 1e-7, same assoc. as reference
          const float den = __fadd_rn(__fsub_rn(__fadd_rn(bArea, ar[j]), inter), 1e-7f);
          const float iou = __fdiv_rn(inter, den);
          if (iou > IOU_T || idx == bestI) rs[j] = NEG_INF;   // includes .at[i].set(-inf)
        }
      }
    }

    // ---------------- emit detection 'det' ----------------
    if (tid == 0) {
      oS[det] = valid ? bestS : 0.f;
      oL[det] = 0.f;                              // labels always 0 (C==1)
      const float4 ob = valid ? bb : make_float4(0.f, 0.f, 0.f, 0.f);
      *(float4*)(oB + det * 4) = ob;
    }
    // reference: ni = where(valid, i, 0); kpts gathered at ni (NOT zeroed!)
    const int kidx = valid ? bestI : 0;
    if (tid < KD) {
      oK[(size_t)det * KD + tid] = gKpt[(size_t)kidx * KD + tid];
    }
    // no extra barrier needed: next iteration's first __syncthreads()
    // orders wS/wI and bSsh/bIsh reuse.
  }
}

extern "C" void kernel_launch(void* const* d_in, const int* in_sizes, int n_in,
                              void* d_out, int out_size, void* d_ws, size_t ws_size,
                              hipStream_t stream) {
  const float* boxes  = (const float*)d_in[0];   // [B, 8400, 4]
  const float* scores = (const float*)d_in[1];   // [B, 8400, 1]
  const float* kpts   = (const float*)d_in[2];   // [B, 8400, 51]

  const int B = in_sizes[0] / (N_ANCH * 4);

  float* out = (float*)d_out;                    // concatenated: nb, ns, nl, kpts
  float* oB = out;
  float* oS = oB + (size_t)B * MAX_DET * 4;
  float* oL = oS + (size_t)B * MAX_DET;
  float* oK = oL + (size_t)B * MAX_DET;

  nms_pose_kernel<<<B, BLOCK, 0, stream>>>(boxes, scores, kpts, oB, oS, oL, oK);
}